// TopKGate_1726576853876
// MI455X (gfx1250) — compile-verified
//
#include <hip/hip_runtime.h>
#include <hip/hip_bf16.h>

// Problem constants (from reference)
#define S_TOK 8192
#define DDIM  2048
#define NEXP  64
#define CAP   128          // ceil(S/E * 1.0) = 128 >= MIN_CAPACITY
#define NBLK  (S_TOK / 16) // 512 GEMM blocks (16 token rows each)
#define LDS_STRIDE 68      // 64 + pad

typedef __attribute__((ext_vector_type(2))) float v2f;
typedef __attribute__((ext_vector_type(8))) float v8f;

// ---------------------------------------------------------------------------
// Kernel 1: logits GEMM (fp32 WMMA 16x16x4) + softmax + argmax per token,
// plus per-block expert histogram and per-token local rank (token order).
// Grid: 512 blocks, 128 threads (4 waves). Wave w owns experts [16w, 16w+16).
// ---------------------------------------------------------------------------
__global__ __launch_bounds__(128) void gate_gemm_kernel(
    const float* __restrict__ x,          // [S, D]
    const float* __restrict__ wg,         // [E, D]
    float* __restrict__ gate_val,         // [S] top-1 gate probability
    int*   __restrict__ expert_idx,       // [S] argmax expert
    int*   __restrict__ local_rank,       // [S] rank within block, token order
    int*   __restrict__ hist,             // [NBLK, E] per-block expert counts
    float* __restrict__ me_part)          // [NBLK, E] per-block gate sums
{
    __shared__ float lds[16 * LDS_STRIDE];
    __shared__ int   sh_e[16];

    const int tid  = threadIdx.x;
    const int wave = tid >> 5;
    const int lane = tid & 31;
    const int m0   = blockIdx.x * 16;     // token row base
    const int n0   = wave * 16;           // expert col base
    const int l15  = lane & 15;
    const int khalf = (lane >> 4) * 2;    // lanes 16-31 hold K=2,3

    // A frag: row m0+l15, K pair starting at khalf (16x4 fp32 layout)
    const float* ap = x  + (size_t)(m0 + l15) * DDIM + khalf;
    // B frag: B[k][n] = wg[n][k] (logits = x @ wg^T), col n0+l15
    const float* bp = wg + (size_t)(n0 + l15) * DDIM + khalf;

    v8f c = {0.f, 0.f, 0.f, 0.f, 0.f, 0.f, 0.f, 0.f};

    #pragma unroll 8
    for (int k0 = 0; k0 < DDIM; k0 += 4) {
        v2f a = *(const v2f*)(ap + k0);   // 8B-aligned: k0%4==0, khalf even
        v2f b = *(const v2f*)(bp + k0);
        c = __builtin_amdgcn_wmma_f32_16x16x4_f32(
                /*neg_a=*/false, a, /*neg_b=*/false, b,
                /*c_mod=*/(short)0, c, /*reuse_a=*/false, /*reuse_b=*/false);
    }

    // C/D 16x16 fp32 layout: VGPR v -> M = v + (lane<16 ? 0 : 8), N = lane&15
    const int rbase = (lane < 16) ? 0 : 8;
    #pragma unroll
    for (int v = 0; v < 8; ++v) {
        lds[(rbase + v) * LDS_STRIDE + n0 + l15] = c[v];
    }
    __syncthreads();

    // Per-row softmax + argmax (threads 0..15, one token row each)
    if (tid < 16) {
        const int r = tid;
        float mx = lds[r * LDS_STRIDE + 0];
        int imax = 0;
        #pragma unroll 4
        for (int e = 1; e < NEXP; ++e) {
            float l = lds[r * LDS_STRIDE + e];
            if (l > mx) { mx = l; imax = e; }   // strict >: first-max like argmax
        }
        float sum = 0.f;
        #pragma unroll 4
        for (int e = 0; e < NEXP; ++e) sum += expf(lds[r * LDS_STRIDE + e] - mx);
        const float inv = 1.0f / sum;
        #pragma unroll 4
        for (int e = 0; e < NEXP; ++e)
            lds[r * LDS_STRIDE + e] = expf(lds[r * LDS_STRIDE + e] - mx) * inv;

        const int s = m0 + r;
        gate_val[s]   = inv;      // gates[s][argmax] = exp(0) * inv
        expert_idx[s] = imax;
        sh_e[r]       = imax;
    }
    __syncthreads();

    // Local rank (token order) within this block's 16 tokens
    if (tid < 16) {
        const int my = sh_e[tid];
        int lr = 0;
        #pragma unroll
        for (int r2 = 0; r2 < 16; ++r2) lr += (r2 < tid && sh_e[r2] == my) ? 1 : 0;
        local_rank[m0 + tid] = lr;
    }

    // Per-block expert histogram + gate column sums (fixed-order downstream)
    if (tid < NEXP) {
        int h = 0;
        #pragma unroll
        for (int r = 0; r < 16; ++r) h += (sh_e[r] == tid) ? 1 : 0;
        hist[(size_t)blockIdx.x * NEXP + tid] = h;

        float s = 0.f;
        #pragma unroll
        for (int r = 0; r < 16; ++r) s += lds[r * LDS_STRIDE + tid];
        me_part[(size_t)blockIdx.x * NEXP + tid] = s;
    }
}

// ---------------------------------------------------------------------------
// Kernel 2: exclusive prefix over block histograms, one expert per thread.
// 512 iterations/thread (vs 8192 serial before); deterministic, no atomics.
// Also produces total pre-capacity counts cnt[e].
// ---------------------------------------------------------------------------
__global__ __launch_bounds__(64) void prefix_kernel(
    const int* __restrict__ hist, int* __restrict__ offs, int* __restrict__ cnt)
{
    const int e = threadIdx.x;
    int c = 0;
    for (int b = 0; b < NBLK; ++b) {
        offs[(size_t)b * NEXP + e] = c;
        c += hist[(size_t)b * NEXP + e];
    }
    cnt[e] = c;
}

// ---------------------------------------------------------------------------
// Kernel 3: l_aux (fixed-order sums -> deterministic) + exp_counts tail.
// ---------------------------------------------------------------------------
__global__ __launch_bounds__(64) void finalize_kernel(
    const float* __restrict__ me_part, const int* __restrict__ cnt,
    float* __restrict__ d_out)
{
    __shared__ float red[64];
    const int e = threadIdx.x;

    float me = 0.f;
    for (int b = 0; b < NBLK; ++b) me += me_part[(size_t)b * NEXP + e];
    me *= (1.0f / (float)S_TOK);
    const float ce = (float)cnt[e] * (1.0f / (float)S_TOK);
    red[e] = me * ce;
    __syncthreads();
    for (int off = 32; off > 0; off >>= 1) {
        if (e < off) red[e] += red[e + off];
        __syncthreads();
    }
    if (e == 0) d_out[0] = red[0] * (float)NEXP;

    // exp_counts (int -> float) at the tail of d_out
    d_out[1 + 2 * (size_t)S_TOK * NEXP * CAP + e] = (float)cnt[e];
}

// ---------------------------------------------------------------------------
// Kernel 4: sparse scatter of combine_weights / dispatch_mask. d_out was
// zeroed by hipMemsetAsync; only kept tokens (rank < CAP) write one entry.
// rank = exclusive block offset + local (token-order) rank, matching
// cumsum(new_mask,axis=0)-1 slot semantics for the kept set.
// ---------------------------------------------------------------------------
__global__ __launch_bounds__(256) void scatter_kernel(
    const float* __restrict__ gate_val, const int* __restrict__ expert_idx,
    const int* __restrict__ local_rank, const int* __restrict__ offs,
    float* __restrict__ d_out)
{
    const int s = blockIdx.x * 256 + threadIdx.x;
    const int e = expert_idx[s];
    const int r = offs[(size_t)(s >> 4) * NEXP + e] + local_rank[s];
    if (r < CAP) {
        const size_t idx = (size_t)s * NEXP * CAP + (size_t)e * CAP + (size_t)r;
        d_out[1 + idx] = gate_val[s];                                  // combine_weights
        d_out[1 + (size_t)S_TOK * NEXP * CAP + idx] = 1.0f;            // dispatch_mask
    }
}

// ---------------------------------------------------------------------------
// d_out layout (reference return order, flat fp32):
//   [0]                       l_aux
//   [1, 1+SEC)                combine_weights  (S,E,C)
//   [1+SEC, 1+2*SEC)          dispatch_mask    (S,E,C)
//   [1+2*SEC, 1+2*SEC+E)      exp_counts
// ---------------------------------------------------------------------------
extern "C" void kernel_launch(void* const* d_in, const int* in_sizes, int n_in,
                              void* d_out, int out_size, void* d_ws, size_t ws_size,
                              hipStream_t stream) {
    const float* x  = (const float*)d_in[0];   // [S, D] fp32
    const float* wg = (const float*)d_in[1];   // [E, D] fp32
    float* out = (float*)d_out;

    // Workspace carve-up (all 256B-aligned offsets)
    char* ws = (char*)d_ws;
    float* gate_val   = (float*)(ws);                  // S*4        = 32768
    int*   expert_idx = (int*)  (ws + 32768);          // S*4        = 32768
    int*   local_rank = (int*)  (ws + 65536);          // S*4        = 32768
    int*   hist       = (int*)  (ws + 98304);          // NBLK*E*4   = 131072
    int*   offs       = (int*)  (ws + 229376);         // NBLK*E*4   = 131072
    int*   cnt        = (int*)  (ws + 360448);         // E*4        = 256
    float* me_part    = (float*)(ws + 360704);         // NBLK*E*4   = 131072

    // Zero the 537 MB output: the unavoidable write floor (~23 us @ 23.3 TB/s).
    hipMemsetAsync(d_out, 0, (size_t)out_size * sizeof(float), stream);

    gate_gemm_kernel<<<NBLK, 128, 0, stream>>>(x, wg, gate_val, expert_idx,
                                               local_rank, hist, me_part);
    prefix_kernel<<<1, 64, 0, stream>>>(hist, offs, cnt);
    finalize_kernel<<<1, 64, 0, stream>>>(me_part, cnt, out);
    scatter_kernel<<<S_TOK / 256, 256, 0, stream>>>(gate_val, expert_idx,
                                                    local_rank, offs, out);
}